// _RestormerBlock_61220463837765
// MI455X (gfx1250) — compile-verified
//
#include <hip/hip_runtime.h>
#include <hip/hip_bf16.h>
#include <math.h>

typedef __attribute__((ext_vector_type(2))) float v2f;
typedef __attribute__((ext_vector_type(4))) float v4f;
typedef __attribute__((ext_vector_type(8))) float v8f;

__device__ __forceinline__ v8f wmma_f32_k4(v2f a, v2f b, v8f c) {
  // D = A(16x4,f32) * B(4x16,f32) + C(16x16,f32)
  return __builtin_amdgcn_wmma_f32_16x16x4_f32(false, a, false, b, (short)0, c,
                                               false, false);
}

// ---------------------------------------------------------------------------
// Channel-wise LayerNorm over dim=1 of (B,C,N). One thread per 4 pixels
// (vectorized b128 loads/stores); consecutive threads -> consecutive n.
// ---------------------------------------------------------------------------
__global__ __launch_bounds__(256) void ln2d_kernel(
    const float* __restrict__ X, const float* __restrict__ g,
    const float* __restrict__ be, float* __restrict__ Out, int C, int N) {
  const size_t idx = (size_t)blockIdx.x * 256 + threadIdx.x;  // over B*N/4
  const int nq = N >> 2;
  const int b = (int)(idx / nq);
  const int n4 = (int)(idx % nq) << 2;
  const float* xp = X + (size_t)b * C * N + n4;
  v4f s = {}, ss = {};
  for (int c = 0; c < C; ++c) {
    v4f v = *reinterpret_cast<const v4f*>(xp + (size_t)c * N);
    s += v;
    ss += v * v;
  }
  const float rc = 1.0f / (float)C;
  v4f mean = s * rc;
  v4f var = ss * rc - mean * mean;  // population variance
  v4f rstd;
#pragma unroll
  for (int i = 0; i < 4; ++i) rstd[i] = rsqrtf(var[i] + 1e-6f);
  float* op = Out + (size_t)b * C * N + n4;
  for (int c = 0; c < C; ++c) {
    v4f v = *reinterpret_cast<const v4f*>(xp + (size_t)c * N);
    *reinterpret_cast<v4f*>(op + (size_t)c * N) = (v - mean) * rstd * g[c] + be[c];
  }
}

// ---------------------------------------------------------------------------
// 1x1 conv as GEMM: Out[b,m,n] = sum_k Wm[m,k] * Y[b,k,n] (+ Res[b,m,n]).
// Block: 256 thr = 8 waves; wave w owns 16x16 tile at (m0, n0 + 16w).
// W tile staged into LDS with ASYNC global->LDS DMA (padded stride to avoid
// bank degeneracy: 192 % 64banks == 0). Y streamed from global (L2 resident)
// with speculative global_prefetch ahead of the K walk.
// ---------------------------------------------------------------------------
#define KPAD 386  // max K (384) + 2
__global__ __launch_bounds__(256) void gemm_wmma_kernel(
    const float* __restrict__ Wm, const float* __restrict__ Y,
    const float* Res, float* Out, int M, int K, int N) {
  __shared__ float sW[16 * KPAD];
  const int b = blockIdx.z;
  const int m0 = blockIdx.y * 16;
  const int tid = threadIdx.x;

  // Asynchronously stage W rows [m0, m0+16) x K into LDS (ASYNCcnt path).
  {
    const unsigned lds0 = (unsigned)(size_t)&sW[0];
    for (int i = tid; i < 16 * K; i += 256) {
      const int r = i / K, c = i - r * K;
      const unsigned laddr = lds0 + (unsigned)((r * KPAD + c) << 2);
      const float* gp = Wm + (size_t)(m0 + r) * K + c;
      asm volatile("global_load_async_to_lds_b32 %0, %1, off"
                   :: "v"(laddr), "v"(gp)
                   : "memory");
    }
    asm volatile("s_wait_asynccnt 0" ::: "memory");
  }
  __syncthreads();

  const int wave = tid >> 5;
  const int lane = tid & 31;
  const int hh = lane >> 4;    // half-wave select (K pairing)
  const int lrow = lane & 15;  // row (A/D) or column (B/D) index
  const int col = blockIdx.x * 128 + wave * 16 + lrow;

  // B-frag pointer walks rows 2*hh, 2*hh+1 then += 4 rows per step.
  const float* yp = Y + (size_t)b * K * N + (size_t)(2 * hh) * N + col;
  const size_t rowstep = (size_t)4 * N;

  v8f acc = {};
#pragma unroll 8
  for (int k = 0; k < K; k += 4) {
    v2f a = *reinterpret_cast<const v2f*>(&sW[lrow * KPAD + k + 2 * hh]);
    v2f bb;
    bb.x = yp[0];
    bb.y = yp[N];
    __builtin_prefetch(yp + 2 * rowstep, 0, 1);  // speculative, branch-free
    acc = wmma_f32_k4(a, bb, acc);
    yp += rowstep;
  }
  // D layout: VGPR r -> row m0 + 8*hh + r, col = n
  float* op = Out + ((size_t)b * M + m0 + 8 * hh) * N + col;
  if (Res) {
    const float* rp = Res + ((size_t)b * M + m0 + 8 * hh) * N + col;
#pragma unroll
    for (int r = 0; r < 8; ++r) op[(size_t)r * N] = acc[r] + rp[(size_t)r * N];
  } else {
#pragma unroll
    for (int r = 0; r < 8; ++r) op[(size_t)r * N] = acc[r];
  }
}

// ---------------------------------------------------------------------------
// Depthwise 3x3, SAME padding, NCHW. Block = one 16x16 output tile of one
// (b,c) plane; 18x18 halo staged in LDS.
// ---------------------------------------------------------------------------
__global__ __launch_bounds__(256) void dwconv3x3_kernel(
    const float* __restrict__ X, const float* __restrict__ Wt,
    float* __restrict__ Out, int C, int H, int W) {
  __shared__ float tile[18 * 18];
  const int tx = threadIdx.x & 15, ty = threadIdx.x >> 4;
  const int tilesx = W >> 4, tilesy = H >> 4;
  int t = blockIdx.x;
  const int bx = t % tilesx;
  t /= tilesx;
  const int by = t % tilesy;
  const int bc = t / tilesy;  // b*C + c
  const int c = bc % C;
  const float* xp = X + (size_t)bc * H * W;
  const int x0 = bx * 16 - 1, y0 = by * 16 - 1;
  for (int i = threadIdx.x; i < 18 * 18; i += 256) {
    const int lx = i % 18, ly = i / 18;
    const int gx = x0 + lx, gy = y0 + ly;
    float v = 0.f;
    if (gx >= 0 && gx < W && gy >= 0 && gy < H) v = xp[(size_t)gy * W + gx];
    tile[i] = v;
  }
  __syncthreads();
  const float* wp = Wt + (size_t)c * 9;
  float acc = 0.f;
#pragma unroll
  for (int dy = 0; dy < 3; ++dy)
#pragma unroll
    for (int dx = 0; dx < 3; ++dx)
      acc += tile[(ty + dy) * 18 + tx + dx] * wp[dy * 3 + dx];
  Out[(size_t)bc * H * W + (size_t)(by * 16 + ty) * W + bx * 16 + tx] = acc;
}

// ---------------------------------------------------------------------------
// 1/max(||row||, eps) for q (ch 0..191) and k (ch 192..383) rows of (B,576,N).
// ---------------------------------------------------------------------------
__global__ __launch_bounds__(256) void l2inv_kernel(const float* __restrict__ QKV,
                                                    float* __restrict__ inv, int N) {
  const int r = blockIdx.x;  // b*384 + ch
  const int b = r / 384, ch = r - b * 384;
  const float* p = QKV + ((size_t)b * 576 + ch) * N;
  float s = 0.f;
  for (int n = threadIdx.x * 4; n < N; n += 1024) {
    v4f v = *reinterpret_cast<const v4f*>(p + n);
    s += v.x * v.x + v.y * v.y + v.z * v.z + v.w * v.w;
  }
  __shared__ float red[256];
  red[threadIdx.x] = s;
  __syncthreads();
  for (int off = 128; off > 0; off >>= 1) {
    if (threadIdx.x < off) red[threadIdx.x] += red[threadIdx.x + off];
    __syncthreads();
  }
  if (threadIdx.x == 0) inv[r] = 1.0f / fmaxf(sqrtf(red[0]), 1e-12f);
}

// ---------------------------------------------------------------------------
// attn_raw += q-chunk @ k-chunk^T : 48x48 per (b,h). K = 16384 split over
// grid.z chunks; partials merged with global_atomic_add_f32. Scaling by
// temperature / inverse norms is folded into the softmax kernel.
// ---------------------------------------------------------------------------
__global__ __launch_bounds__(32) void attn_qk_kernel(
    const float* __restrict__ QKV, float* __restrict__ Attn, int N, int chunk) {
  const int tile = blockIdx.x;  // 0..8
  const int bh = blockIdx.y;    // 0..31
  const int b = bh >> 2, h = bh & 3;
  const int i0 = (tile / 3) * 16, j0 = (tile % 3) * 16;
  const int lane = threadIdx.x & 31;
  const int hh = lane >> 4, lrow = lane & 15;
  const int n0 = blockIdx.z * chunk;
  const float* qp = QKV + ((size_t)b * 576 + h * 48 + i0 + lrow) * N + n0 + 2 * hh;
  const float* kp =
      QKV + ((size_t)b * 576 + 192 + h * 48 + j0 + lrow) * N + n0 + 2 * hh;
  v8f acc = {};
#pragma unroll 8
  for (int n = 0; n < chunk; n += 4) {
    v2f a = *reinterpret_cast<const v2f*>(qp);
    v2f bb = *reinterpret_cast<const v2f*>(kp);
    acc = wmma_f32_k4(a, bb, acc);
    qp += 4;
    kp += 4;
  }
  const int j = j0 + lrow;
#pragma unroll
  for (int r = 0; r < 8; ++r) {
    const int i = i0 + 8 * hh + r;
    atomicAdd(&Attn[((size_t)bh * 48 + i) * 48 + j], acc[r]);
  }
}

// ---------------------------------------------------------------------------
// Scale by temperature * invq * invk, then row softmax over 48. One thread
// per row (row = bh*48 + i).
// ---------------------------------------------------------------------------
__global__ __launch_bounds__(256) void softmax48_kernel(
    float* __restrict__ A, const float* __restrict__ inv,
    const float* __restrict__ temp, int rows) {
  const int r = blockIdx.x * 256 + threadIdx.x;
  if (r >= rows) return;
  const int bh = r / 48, i = r - bh * 48;
  const int b = bh >> 2, h = bh & 3;
  const float tiq = temp[h] * inv[b * 384 + h * 48 + i];
  const float* ik = inv + b * 384 + 192 + h * 48;
  float* p = A + (size_t)r * 48;
  float mx = -INFINITY;
  for (int j = 0; j < 48; ++j) {
    const float v = p[j] * tiq * ik[j];
    p[j] = v;
    mx = fmaxf(mx, v);
  }
  float s = 0.f;
  for (int j = 0; j < 48; ++j) s += __expf(p[j] - mx);
  const float rs = 1.0f / s;
  for (int j = 0; j < 48; ++j) p[j] = __expf(p[j] - mx) * rs;
}

// ---------------------------------------------------------------------------
// out = attn(48x48) @ v(48xN) per (b,h). M tiles = 3, K = 48 (12 WMMA steps).
// ---------------------------------------------------------------------------
__global__ __launch_bounds__(256) void attn_v_kernel(
    const float* __restrict__ Attn, const float* __restrict__ QKV,
    float* __restrict__ Out, int N) {
  const int bh = blockIdx.z;
  const int b = bh >> 2, h = bh & 3;
  const int m0 = blockIdx.y * 16;
  const int wave = threadIdx.x >> 5, lane = threadIdx.x & 31;
  const int hh = lane >> 4, lrow = lane & 15;
  const int col = blockIdx.x * 128 + wave * 16 + lrow;
  const float* A = Attn + (size_t)bh * 48 * 48 + (size_t)(m0 + lrow) * 48 + 2 * hh;
  const float* vp = QKV + ((size_t)b * 576 + 384 + h * 48 + 2 * hh) * N + col;
  const size_t rowstep = (size_t)4 * N;
  v8f acc = {};
#pragma unroll
  for (int k = 0; k < 48; k += 4) {
    v2f a = *reinterpret_cast<const v2f*>(A + k);
    v2f bb;
    bb.x = vp[0];
    bb.y = vp[N];
    acc = wmma_f32_k4(a, bb, acc);
    vp += rowstep;
  }
  float* op = Out + ((size_t)b * 192 + h * 48 + m0 + 8 * hh) * N + col;
#pragma unroll
  for (int r = 0; r < 8; ++r) op[(size_t)r * N] = acc[r];
}

// ---------------------------------------------------------------------------
// GDFN gate: out[b,c,n] = gelu_exact(y[b,c,n]) * y[b,c+384,n], vectorized x4.
// ---------------------------------------------------------------------------
__global__ __launch_bounds__(256) void gelu_gate_kernel(
    const float* __restrict__ Yh, float* __restrict__ Out, int N) {
  const size_t idx = (size_t)blockIdx.x * 256 + threadIdx.x;  // over B*384*N/4
  const int nq = N >> 2;
  const int n4 = (int)(idx % nq) << 2;
  size_t t = idx / nq;
  const int c = (int)(t % 384);
  const int b = (int)(t / 384);
  const float* base = Yh + (size_t)b * 768 * N;
  v4f v1 = *reinterpret_cast<const v4f*>(base + (size_t)c * N + n4);
  v4f v2 = *reinterpret_cast<const v4f*>(base + (size_t)(c + 384) * N + n4);
  v4f o;
#pragma unroll
  for (int i = 0; i < 4; ++i)
    o[i] = 0.5f * v1[i] * (1.0f + erff(v1[i] * 0.7071067811865475f)) * v2[i];
  *reinterpret_cast<v4f*>(Out + (((size_t)b * 384 + c) * N + n4)) = o;
}

// ---------------------------------------------------------------------------
extern "C" void kernel_launch(void* const* d_in, const int* in_sizes, int n_in,
                              void* d_out, int out_size, void* d_ws, size_t ws_size,
                              hipStream_t stream) {
  const float* x       = (const float*)d_in[0];
  const float* n1w     = (const float*)d_in[1];
  const float* n1b     = (const float*)d_in[2];
  const float* qkv_w   = (const float*)d_in[3];
  const float* qkv_dw  = (const float*)d_in[4];
  const float* temp    = (const float*)d_in[5];
  const float* attn_ow = (const float*)d_in[6];
  const float* n2w     = (const float*)d_in[7];
  const float* n2b     = (const float*)d_in[8];
  const float* ffn_iw  = (const float*)d_in[9];
  const float* ffn_dw  = (const float*)d_in[10];
  const float* ffn_ow  = (const float*)d_in[11];
  float* out = (float*)d_out;
  float* ws = (float*)d_ws;

  const int B = 8, C = 192, H = 128, Wd = 128, NH = 4;
  const int N = H * Wd;                     // 16384
  const size_t F_BIG = (size_t)768 * B * N; // 100,663,296 floats
  float* bufA = ws;                 // 768ch scratch (also attn-out 192ch alias)
  float* bufB = ws + F_BIG;         // 768ch scratch
  float* bufY = ws + 2 * F_BIG;     // LN output (B,192,N)
  float* invn = bufY + (size_t)C * B * N;  // (B,384)
  float* attn = invn + (size_t)B * 384;    // (B,4,48,48)

  // ---- MDTA ----
  ln2d_kernel<<<dim3((B * N / 4) / 256), 256, 0, stream>>>(x, n1w, n1b, bufY, C, N);
  gemm_wmma_kernel<<<dim3(N / 128, 576 / 16, B), 256, 0, stream>>>(
      qkv_w, bufY, nullptr, bufA, 576, 192, N);
  dwconv3x3_kernel<<<dim3(B * 576 * (H / 16) * (Wd / 16)), 256, 0, stream>>>(
      bufA, qkv_dw, bufB, 576, H, Wd);
  l2inv_kernel<<<dim3(B * 384), 256, 0, stream>>>(bufB, invn, N);
  hipMemsetAsync(attn, 0, sizeof(float) * B * NH * 48 * 48, stream);
  attn_qk_kernel<<<dim3(9, B * NH, 16), 32, 0, stream>>>(bufB, attn, N, N / 16);
  softmax48_kernel<<<dim3(6), 256, 0, stream>>>(attn, invn, temp, B * NH * 48);
  attn_v_kernel<<<dim3(N / 128, 3, B * NH), 256, 0, stream>>>(attn, bufB, bufA, N);
  gemm_wmma_kernel<<<dim3(N / 128, 192 / 16, B), 256, 0, stream>>>(
      attn_ow, bufA, x, out, 192, 192, N);  // + residual x

  // ---- GDFN ----
  ln2d_kernel<<<dim3((B * N / 4) / 256), 256, 0, stream>>>(out, n2w, n2b, bufY, C, N);
  gemm_wmma_kernel<<<dim3(N / 128, 768 / 16, B), 256, 0, stream>>>(
      ffn_iw, bufY, nullptr, bufB, 768, 192, N);
  dwconv3x3_kernel<<<dim3(B * 768 * (H / 16) * (Wd / 16)), 256, 0, stream>>>(
      bufB, ffn_dw, bufA, 768, H, Wd);
  gelu_gate_kernel<<<dim3((unsigned)(((size_t)B * 384 * N / 4) / 256)), 256, 0,
                     stream>>>(bufA, bufB, N);
  gemm_wmma_kernel<<<dim3(N / 128, 192 / 16, B), 256, 0, stream>>>(
      ffn_ow, bufB, out, out, 192, 384, N);  // + residual (in-place)
}